// SpectralConv1d_979252544384
// MI455X (gfx1250) — compile-verified
//
#include <hip/hip_runtime.h>
#include <cmath>

typedef float v2f __attribute__((ext_vector_type(2)));
typedef float v8f __attribute__((ext_vector_type(8)));

#define N_TOT   8192
#define NMODE   16
#define IND     64
#define OUTD    64
#define NROWS   4096          // B * IN_DIM
#define NCHUNK  16
#define KC      512           // K per forward chunk
#define NIT     (KC/32)       // 16 pipeline iterations per wave
#define LDS_STRIDE 36         // padded row stride (floats) -> conflict-free ds_load_b64

// workspace layout (float offsets)
#define OFF_BF  0                           // basisF_pack: 2048 kgroups * 32 lanes * 4
#define SZ_BF   (2048*32*4)
#define OFF_BI  (OFF_BF + SZ_BF)            // basisI_pack: 512 ntiles * 8 ksteps * 32 lanes * 2
#define SZ_BI   (512*8*32*2)
#define OFF_XR  (OFF_BI + SZ_BI)            // forward partials (real): NCHUNK*4096*16
#define SZ_X    (NCHUNK*NROWS*NMODE)
#define OFF_XI  (OFF_XR + SZ_X)             // forward partials (imag)
#define OFF_UP  (OFF_XI + SZ_X)             // Upack (A-operand layout): 256*8*32*2

#define TWO_PI_OVER_N  (6.283185307179586f / 8192.0f)
#define INV_SQRT_N     0.011048543456039806f   // 1/sqrt(8192)

// ---------------- basis precompute ----------------
// Forward basis, packed per WMMA B-operand lane layout.
// group g covers K rows n0=4g+2*(lane>>4), n0+1 ; N column = mode m = lane&15.
// floats: {cos(n0,m), cos(n1,m), -sin(n0,m), -sin(n1,m)} * 1/sqrt(N)
__global__ void k_initF(float* __restrict__ bf) {
    int g = blockIdx.x;          // 0..2047
    int L = threadIdx.x;         // 0..31
    int n0 = 4*g + 2*(L >> 4);
    int m  = L & 15;
    float th0 = TWO_PI_OVER_N * (float)((m * n0)     & (N_TOT - 1));
    float th1 = TWO_PI_OVER_N * (float)((m * (n0+1)) & (N_TOT - 1));
    float4 v;
    v.x =  cosf(th0) * INV_SQRT_N;
    v.y =  cosf(th1) * INV_SQRT_N;
    v.z = -sinf(th0) * INV_SQRT_N;
    v.w = -sinf(th1) * INV_SQRT_N;
    reinterpret_cast<float4*>(bf)[g*32 + L] = v;
}

// Inverse basis V(32 x 8192): rows 0..15 = cos(2*pi*k*n/N), rows 16..31 = sin.
// Packed per B-operand layout per (ntile, kstep s): lane holds V[4s+2*(L>>4)+t][nt*16+(L&15)], t=0,1
__global__ void k_initI(float* __restrict__ bi) {
    int idx = blockIdx.x * blockDim.x + threadIdx.x;   // 0..131071
    int nt = idx >> 8;
    int s  = (idx >> 5) & 7;
    int L  = idx & 31;
    int n  = nt*16 + (L & 15);
    float o2[2];
    #pragma unroll
    for (int t = 0; t < 2; ++t) {
        int K = 4*s + 2*(L >> 4) + t;
        int k = (K < 16) ? K : (K - 16);
        float th = TWO_PI_OVER_N * (float)((k * n) & (N_TOT - 1));
        o2[t] = (K < 16) ? cosf(th) : sinf(th);
    }
    float2 v; v.x = o2[0]; v.y = o2[1];
    reinterpret_cast<float2*>(bi)[(nt*8 + s)*32 + L] = v;
}

// ---------------- stage 1: forward truncated DFT ----------------
// C(4096x32) = x(4096x8192) . F(8192x32), K split 16 ways, partials to ws.
// Barrier-free: each wave owns a private LDS region; DS ops are in-order per
// wave, so ds_store -> ds_load visibility is guaranteed. Next tile's global
// loads are issued into staging VGPRs before the current tile's compute, so
// HBM latency overlaps the 16 WMMAs + 8 ds_loads.
__global__ __launch_bounds__(256) void k_fwd(const float* __restrict__ x,
                                             const float* __restrict__ bf,
                                             float* __restrict__ xr,
                                             float* __restrict__ xi) {
    __shared__ float lds[8][16 * LDS_STRIDE];
    int warp = threadIdx.x >> 5;
    int lane = threadIdx.x & 31;
    int wid   = blockIdx.x * 8 + warp;   // 0..4095
    int rb    = wid >> 4;                // 0..255  (16-row block)
    int chunk = wid & 15;                // 0..15   (K chunk)
    int kbase = chunk * KC;
    float* my = &lds[warp][0];
    const float* xrow0 = x + (size_t)rb * 16 * N_TOT + kbase;

    v8f accC = {0,0,0,0,0,0,0,0};
    v8f accS = {0,0,0,0,0,0,0,0};
    int M  = lane & 15;
    int K0 = (lane >> 4) * 2;

    // per-lane staging-load geometry (16 rows x 32 cols per tile)
    int r_[4], c_[4];
    #pragma unroll
    for (int t = 0; t < 4; ++t) {
        int idx = t*32 + lane;
        r_[t] = idx >> 3;
        c_[t] = (idx & 7) * 4;
    }

    // prologue: stage tile 0 into registers
    float4 st[4];
    #pragma unroll
    for (int t = 0; t < 4; ++t)
        st[t] = *reinterpret_cast<const float4*>(xrow0 + (size_t)r_[t] * N_TOT + c_[t]);

    for (int it = 0; it < NIT; ++it) {
        // commit staged tile to LDS (s_wait_loadcnt lands here)
        #pragma unroll
        for (int t = 0; t < 4; ++t)
            *reinterpret_cast<float4*>(&my[r_[t] * LDS_STRIDE + c_[t]]) = st[t];

        // issue next tile's global loads (clamped on last iter: harmless L2 hit)
        int itn = (it + 1 < NIT) ? (it + 1) : it;
        const float* xt = xrow0 + itn * 32;
        #pragma unroll
        for (int t = 0; t < 4; ++t)
            st[t] = *reinterpret_cast<const float4*>(xt + (size_t)r_[t] * N_TOT + c_[t]);

        // compute current tile: 8 k-steps x 2 WMMA (cos + sin share A)
        const float4* bgrp = reinterpret_cast<const float4*>(bf)
                             + ((size_t)(kbase + it * 32) >> 2) * 32 + lane;
        #pragma unroll
        for (int kk = 0; kk < 32; kk += 4) {
            v2f a = *reinterpret_cast<const v2f*>(&my[M * LDS_STRIDE + kk + K0]);
            float4 b4 = bgrp[(kk >> 2) * 32];
            v2f bc; bc.x = b4.x; bc.y = b4.y;
            v2f bs; bs.x = b4.z; bs.y = b4.w;
            accC = __builtin_amdgcn_wmma_f32_16x16x4_f32(false, a, false, bc, (short)0, accC, false, false);
            accS = __builtin_amdgcn_wmma_f32_16x16x4_f32(false, a, false, bs, (short)0, accS, false, false);
        }
    }

    // store partial C tiles in plain (row, mode) layout
    #pragma unroll
    for (int v = 0; v < 8; ++v) {
        int Mo   = v + 8 * (lane >> 4);
        int row  = rb * 16 + Mo;
        int mode = lane & 15;
        size_t off = ((size_t)chunk * NROWS + row) * NMODE + mode;
        xr[off] = accC[v];
        xi[off] = accS[v];
    }
}

// ---------------- stage 2: chunk-reduce + complex mode mix ----------------
// block = (b, k); 64 threads over o. Writes U pre-swizzled into A-operand layout.
__global__ __launch_bounds__(64) void k_mix(const float* __restrict__ xr,
                                            const float* __restrict__ xi,
                                            const float* __restrict__ wgt,
                                            float* __restrict__ up) {
    __shared__ float shr[IND], shi[IND];
    int b = blockIdx.x >> 4;
    int k = blockIdx.x & 15;
    int t = threadIdx.x;
    {
        int row = b * IND + t;
        float sr = 0.f, si = 0.f;
        #pragma unroll
        for (int c = 0; c < NCHUNK; ++c) {
            size_t off = ((size_t)c * NROWS + row) * NMODE + k;
            sr += xr[off]; si += xi[off];
        }
        shr[t] = sr; shi[t] = si;
    }
    __syncthreads();
    int o = t;
    float yr = 0.f, yi = 0.f;
    for (int i = 0; i < IND; ++i) {
        const float* wp = wgt + (((size_t)i * OUTD + o) * NMODE + k) * 2;
        float wr = wp[0], wi = wp[1];
        float ar = shr[i], ai = shi[i];
        yr += ar * wr - ai * wi;
        yi += ar * wi + ai * wr;
    }
    float s  = (k == 0 ? 1.0f : 2.0f) * INV_SQRT_N;
    float Ur =  yr * s;
    float Ui = -yi * s;
    int r  = b * OUTD + o;
    int rb = r >> 4, M = r & 15;
    int   cols[2] = {k, 16 + k};
    float vals[2] = {Ur, Ui};
    #pragma unroll
    for (int j = 0; j < 2; ++j) {
        int cc = cols[j];
        int sg = cc >> 2, rem = cc & 3, half = rem >> 1, pos = rem & 1;
        int L = (half << 4) | M;
        up[(((size_t)rb * 8 + sg) * 32 + L) * 2 + pos] = vals[j];
    }
}

// ---------------- stage 3: truncated inverse DFT ----------------
// out(4096x8192) = U(4096x32) . V(32x8192); one wave owns a 16-row block x 8 col tiles.
__global__ __launch_bounds__(256) void k_inv(const float* __restrict__ up,
                                             const float* __restrict__ bi,
                                             float* __restrict__ out) {
    int warp = threadIdx.x >> 5;
    int lane = threadIdx.x & 31;
    int wid  = blockIdx.x * 8 + warp;  // 0..16383
    int rb   = wid >> 6;               // 0..255
    int span = wid & 63;               // 0..63  -> 8 col tiles each
    v2f aReg[8];
    #pragma unroll
    for (int s = 0; s < 8; ++s)
        aReg[s] = *reinterpret_cast<const v2f*>(up + (((size_t)rb * 8 + s) * 32 + lane) * 2);
    int M16 = lane & 15;
    int hi  = lane >> 4;
    for (int tl = 0; tl < 8; ++tl) {
        int nt = span * 8 + tl;
        v8f acc = {0,0,0,0,0,0,0,0};
        const v2f* bp = reinterpret_cast<const v2f*>(bi) + (size_t)nt * 8 * 32 + lane;
        #pragma unroll
        for (int s = 0; s < 8; ++s) {
            v2f b = bp[s * 32];
            acc = __builtin_amdgcn_wmma_f32_16x16x4_f32(false, aReg[s], false, b, (short)0, acc, false, false);
        }
        int col = nt * 16 + M16;
        #pragma unroll
        for (int v = 0; v < 8; ++v) {
            int row = rb * 16 + v + 8 * hi;
            out[(size_t)row * N_TOT + col] = acc[v];
        }
    }
}

extern "C" void kernel_launch(void* const* d_in, const int* in_sizes, int n_in,
                              void* d_out, int out_size, void* d_ws, size_t ws_size,
                              hipStream_t stream) {
    const float* x = (const float*)d_in[0];
    const float* w = (const float*)d_in[1];
    float* out = (float*)d_out;
    float* ws  = (float*)d_ws;
    float* bf = ws + OFF_BF;
    float* bi = ws + OFF_BI;
    float* xr = ws + OFF_XR;
    float* xi = ws + OFF_XI;
    float* up = ws + OFF_UP;

    k_initF<<<2048, 32, 0, stream>>>(bf);
    k_initI<<<512, 256, 0, stream>>>(bi);
    k_fwd  <<<512, 256, 0, stream>>>(x, bf, xr, xi);
    k_mix  <<<1024, 64, 0, stream>>>(xr, xi, w, up);
    k_inv  <<<2048, 256, 0, stream>>>(up, bi, out);
}